// RewardFunctionER_69423851373231
// MI455X (gfx1250) — compile-verified
//
#include <hip/hip_runtime.h>

// Problem constants (from the reference): S,H,W=256, F=6, B=4096
#define S_DIM 256
#define H_DIM 256
#define W_DIM 256
#define F_DIM 6
#define B_DIM 4096

typedef float v2f __attribute__((ext_vector_type(2)));
typedef float v8f __attribute__((ext_vector_type(8)));

// ds_swizzle group-of-32 XOR pattern: offset = {0, xor[14:10], or[9:5]=0, and[4:0]=0x1f}
template <int XM>
__device__ __forceinline__ float swizzle_xor(float v) {
    return __int_as_float(
        __builtin_amdgcn_ds_swizzle(__float_as_int(v), (XM << 10) | 0x1f));
}

// One wave (32 lanes, wave32) per query q = b*4 + t*2 + which.
// Computes v[q] = sum_s softmax(v)_s * v_s where
//   v_s = dot(succ[s, x, y, 0:6], w)
// using V_WMMA_F32_16X16X4_F32: A = weight rows, B = 16 gathered succ rows.
__global__ __launch_bounds__(256) void rf_softmax_value_kernel(
    const float* __restrict__ phi,    // [B][2][10]
    const float* __restrict__ w_lin,  // [6]
    const float* __restrict__ succ,   // [S][H][W][F]
    float* __restrict__ vout)         // [B*2*2]
{
    const int lane = threadIdx.x & 31;
    const int wave = threadIdx.x >> 5;
    const int q    = blockIdx.x * 8 + wave;   // 0..16383
    const int b    = q >> 2;
    const int t    = (q >> 1) & 1;
    const int wh   = q & 1;                   // 0 = ss, 1 = es

    const float* p = phi + (b * 2 + t) * 10;
    const int x = (int)p[6 + 2 * wh];
    const int y = (int)p[7 + 2 * wh];

    const float w0 = w_lin[0], w1 = w_lin[1], w2 = w_lin[2];
    const float w3 = w_lin[3], w4 = w_lin[4], w5 = w_lin[5];

    const bool hi = (lane >= 16);
    // A-matrix 16x4 layout: lanes 0-15 hold K=0,1 (VGPR0,VGPR1); lanes 16-31 hold K=2,3.
    // All 16 rows identical => every row of D is the same dot product.
    v2f a1, a2;
    a1.x = hi ? w2 : w0;   a1.y = hi ? w3 : w1;   // K = 0..3
    a2.x = hi ? 0.f : w4;  a2.y = hi ? 0.f : w5;  // K = 4,5 (6,7 zero-padded)

    // (x,y) is wave-uniform: force the tile base into an SGPR pair so every
    // gather lowers to GVS mode: global_load_b64 vdst, voff32, s[base].
    const unsigned rowstride = H_DIM * W_DIM * F_DIM;                        // 393216
    unsigned base = (unsigned)x * (W_DIM * F_DIM) + (unsigned)y * F_DIM;
    base = (unsigned)__builtin_amdgcn_readfirstlane((int)base);
    const float* __restrict__ bp = succ + base;

    // B-matrix 4x16: lane n (0-15) = column n with K=0,1; lanes 16-31 = K=2,3.
    const int      n      = lane & 15;
    const unsigned o1base = (unsigned)n * rowstride + (hi ? 2u : 0u); // f0,f1 / f2,f3
    const unsigned o2base = (unsigned)n * rowstride + 4u;             // f4,f5

    float vvals[16];
#pragma unroll
    for (int chunk = 0; chunk < 16; ++chunk) {
        // Single 32-bit per-lane offset (constant per unrolled iteration),
        // applied to the uniform base pointer in one GEP index.
        const unsigned step = (unsigned)chunk * 16u * rowstride;
        const unsigned o1   = o1base + step;
        const unsigned o2   = o2base + step;
        v2f b1 = *(const v2f*)(bp + o1);
        v2f b2 = *(const v2f*)(bp + o2);  // lanes>=16 feed K=6,7 (zero weights)
        v8f c  = {};
        c = __builtin_amdgcn_wmma_f32_16x16x4_f32(false, a1, false, b1,
                                                  (short)0, c, false, false);
        c = __builtin_amdgcn_wmma_f32_16x16x4_f32(false, a2, false, b2,
                                                  (short)0, c, false, false);
        // D row 0 (and every row): c[0] in lane L = v_{chunk*16 + (L&15)}
        vvals[chunk] = c[0];
    }

    // Softmax-weighted mean over 256 values: 16 per lane in lanes 0-15
    // (lanes 16-31 are exact duplicates, so XOR-16 reduction step is skipped).
    float m = vvals[0];
#pragma unroll
    for (int j = 1; j < 16; ++j) m = fmaxf(m, vvals[j]);
    m = fmaxf(m, swizzle_xor<1>(m));
    m = fmaxf(m, swizzle_xor<2>(m));
    m = fmaxf(m, swizzle_xor<4>(m));
    m = fmaxf(m, swizzle_xor<8>(m));

    float se = 0.f, wsum = 0.f;
#pragma unroll
    for (int j = 0; j < 16; ++j) {
        float e = __expf(vvals[j] - m);
        se   += e;
        wsum += e * vvals[j];
    }
    se   += swizzle_xor<1>(se);
    wsum += swizzle_xor<1>(wsum);
    se   += swizzle_xor<2>(se);
    wsum += swizzle_xor<2>(wsum);
    se   += swizzle_xor<4>(se);
    wsum += swizzle_xor<4>(wsum);
    se   += swizzle_xor<8>(se);
    wsum += swizzle_xor<8>(wsum);

    if (lane == 0) vout[q] = wsum / se;
}

// One thread per b: pr dot products + sigmoid combine.
__global__ __launch_bounds__(256) void rf_combine_kernel(
    const float* __restrict__ phi,    // [B][2][10]
    const float* __restrict__ w_lin,  // [6]
    const float* __restrict__ vtmp,   // [B*2*2]
    float* __restrict__ out)          // [B][2]
{
    const int b = blockIdx.x * blockDim.x + threadIdx.x;
    if (b >= B_DIM) return;

    const float* p = phi + b * 20;
    float pr0 = 0.f, pr1 = 0.f;
#pragma unroll
    for (int f = 0; f < F_DIM; ++f) {
        const float w = w_lin[f];
        pr0 += p[f] * w;
        pr1 += p[10 + f] * w;
    }
    const float vss0 = vtmp[b * 4 + 0];
    const float ves0 = vtmp[b * 4 + 1];
    const float vss1 = vtmp[b * 4 + 2];
    const float ves1 = vtmp[b * 4 + 3];

    const float left_der  = pr0 + (ves0 - vss0);
    const float right_der = pr1 + (ves1 - vss1);
    const float d         = left_der - right_der;

    out[b * 2 + 0] = 1.f / (1.f + __expf(-d));
    out[b * 2 + 1] = 1.f / (1.f + __expf(d));
}

extern "C" void kernel_launch(void* const* d_in, const int* in_sizes, int n_in,
                              void* d_out, int out_size, void* d_ws, size_t ws_size,
                              hipStream_t stream) {
    const float* phi   = (const float*)d_in[0];  // (B,2,10)
    const float* w_lin = (const float*)d_in[1];  // (1,6)
    const float* succ  = (const float*)d_in[2];  // (S,H,W,F)
    float*       vtmp  = (float*)d_ws;           // 16384 floats scratch
    float*       out   = (float*)d_out;          // 8192 floats

    // 16384 queries, 1 wave each, 8 waves per 256-thread block
    rf_softmax_value_kernel<<<2048, 256, 0, stream>>>(phi, w_lin, succ, vtmp);
    rf_combine_kernel<<<(B_DIM + 255) / 256, 256, 0, stream>>>(phi, w_lin, vtmp, out);
}